// GeometrySerializer_61864708931987
// MI455X (gfx1250) — compile-verified
//
#include <hip/hip_runtime.h>
#include <hip/hip_bf16.h>

// ---------------------------------------------------------------------------
// GeometrySerializer for MI455X (gfx1250, wave32).
// Memory-bound: 402 MB token gather + ~190 MB hybrid bitonic sort traffic.
// CDNA5-specific path: async global<->LDS DMA (ASYNCcnt) for the row gather.
// ---------------------------------------------------------------------------

#define DEV static __device__ __forceinline__

constexpr int B  = 8;
constexpr int L  = 16384;
constexpr int NI = 8192;
constexpr int N  = L + NI;     // 24576
constexpr int C  = 256;
constexpr int V  = 4;
constexpr int M  = 32768;      // per-batch padded (pow2) sort length
constexpr long long KMULT  = 100000LL;
constexpr long long KMULT2 = KMULT * KMULT;   // 1e10

// ---- CDNA5 async global<->LDS DMA (probe via __has_builtin) ----------------
#if defined(__has_builtin)
#if __has_builtin(__builtin_amdgcn_global_load_async_to_lds_b128) && \
    __has_builtin(__builtin_amdgcn_global_store_async_from_lds_b128)
#define HAVE_ASYNC_LDS 1
#endif
#endif

// Builtin signature (from hipcc diagnostic): param0 = AS1 (global) v4i*,
// param1 = LDS-side v4i* (AS3), then imm offset, imm cpol.
typedef int v4i __attribute__((vector_size(16)));
typedef __attribute__((address_space(1))) v4i* gv4p;   // global
typedef __attribute__((address_space(3))) v4i* lv4p;   // LDS

DEV void wait_asynccnt0() {
#if defined(__has_builtin) && __has_builtin(__builtin_amdgcn_s_wait_asynccnt)
  __builtin_amdgcn_s_wait_asynccnt(0);
#else
  asm volatile("s_wait_asynccnt 0" ::: "memory");
#endif
}

// ---- helpers ---------------------------------------------------------------
DEV unsigned long long smap(long long v) {
  return (unsigned long long)v ^ 0x8000000000000000ULL;   // order-preserving
}
DEV long long sunmap(unsigned long long u) {
  return (long long)(u ^ 0x8000000000000000ULL);
}
DEV bool key_gt(ulonglong2 a, ulonglong2 b) {
  return (a.x > b.x) || (a.x == b.x && a.y > b.y);
}

// Recomputable projection: coords, pad, cam for element i of batch b.
DEV void project_point(int b, int i,
                       const float* __restrict__ lidar_coords,
                       const float* __restrict__ img_coords,
                       const int* __restrict__ lmask,
                       const int* __restrict__ imask,
                       const float* __restrict__ Tinv,
                       const float* __restrict__ Kmat,
                       float c[3], int* pad, long long* cam) {
  if (i < L) {
    const float* p = lidar_coords + ((size_t)b * L + i) * 3;
    float x = p[0], y = p[1], z = p[2];
    const float* Ti = Tinv + b * 16;
    float c0 = Ti[0]  * x + Ti[1]  * y + Ti[2]  * z + Ti[3];
    float c1 = Ti[4]  * x + Ti[5]  * y + Ti[6]  * z + Ti[7];
    float c2 = Ti[8]  * x + Ti[9]  * y + Ti[10] * z + Ti[11];
    const float* Km = Kmat + (size_t)b * V * 9;            // MAIN_VIEW = 0
    float i0 = Km[0] * c0 + Km[1] * c1 + Km[2] * c2;
    float i1 = Km[3] * c0 + Km[4] * c1 + Km[5] * c2;
    float i2 = Km[6] * c0 + Km[7] * c1 + Km[8] * c2;
    float dm = fmaxf(i2, 1e-5f);
    c[0] = 0.0f; c[1] = i0 / dm; c[2] = i1 / dm;
    *pad = (lmask[(size_t)b * L + i] != 0) || !(i2 > 1e-5f);
    *cam = 0;
  } else {
    int j = i - L;
    const float* p = img_coords + ((size_t)b * NI + j) * 3;
    c[0] = p[0]; c[1] = p[1]; c[2] = p[2];
    *pad = (imask[(size_t)b * NI + j] != 0);
    *cam = (long long)p[0];                                // trunc toward 0
  }
}

// ---- kernel 0: per-batch 4x4 inverse + reduction init ----------------------
__global__ void k_init(const float* __restrict__ T_c2w,
                       float* __restrict__ Tinv,
                       unsigned long long* __restrict__ red) {
  int t = threadIdx.x;
  if (t < B * 8) {                 // slots: 0..2 min, 3..4 max, 5 anyvalid
    int s = t & 7;
    red[t] = (s < 3) ? ~0ULL : 0ULL;
  }
  if (t < B) {
    const float* A = T_c2w + (size_t)t * V * 16;           // view 0
    float a[4][8];
    for (int r = 0; r < 4; ++r)
      for (int c = 0; c < 4; ++c) {
        a[r][c]     = A[r * 4 + c];
        a[r][4 + c] = (r == c) ? 1.0f : 0.0f;
      }
    for (int col = 0; col < 4; ++col) {
      int piv = col; float best = fabsf(a[col][col]);
      for (int r = col + 1; r < 4; ++r) {
        float v = fabsf(a[r][col]);
        if (v > best) { best = v; piv = r; }
      }
      if (piv != col)
        for (int c = 0; c < 8; ++c) { float tmp = a[col][c]; a[col][c] = a[piv][c]; a[piv][c] = tmp; }
      float inv = 1.0f / a[col][col];
      for (int c = 0; c < 8; ++c) a[col][c] *= inv;
      for (int r = 0; r < 4; ++r) {
        if (r == col) continue;
        float f = a[r][col];
        for (int c = 0; c < 8; ++c) a[r][c] -= f * a[col][c];
      }
    }
    float* o = Tinv + t * 16;
    for (int r = 0; r < 4; ++r)
      for (int c = 0; c < 4; ++c) o[r * 4 + c] = a[r][4 + c];
  }
}

// ---- kernel 1: project + min/max reduction over valid ----------------------
__global__ void k_reduce(const float* __restrict__ lidar_coords,
                         const float* __restrict__ img_coords,
                         const int* __restrict__ lmask,
                         const int* __restrict__ imask,
                         const float* __restrict__ Tinv,
                         const float* __restrict__ Kmat,
                         unsigned long long* __restrict__ red) {
  int t = blockIdx.x * 256 + threadIdx.x;
  if (t >= B * N) return;
  int b = t / N, i = t % N;
  float c[3]; int pad; long long cam;
  project_point(b, i, lidar_coords, img_coords, lmask, imask, Tinv, Kmat, c, &pad, &cam);
  if (cam == 0 && !pad) {
    long long q0 = (long long)floorf(c[0] / 16.0f);
    long long q1 = (long long)floorf(c[1] / 16.0f);
    long long q2 = (long long)floorf(c[2] / 16.0f);
    unsigned long long* r = red + (size_t)b * 8;
    atomicMin(&r[0], smap(q0));
    atomicMin(&r[1], smap(q1));
    atomicMin(&r[2], smap(q2));
    atomicMax(&r[3], smap(q1));
    atomicMax(&r[4], smap(q2));
    atomicOr(&r[5], 1ULL);
  }
}

// ---- kernel 2: finalize mins / max_d1 / max_d2 (+ scalar L output) ---------
__global__ void k_finalize(const unsigned long long* __restrict__ red,
                           long long* __restrict__ fin,
                           float* __restrict__ out_L) {
  int b = threadIdx.x;
  if (b == 0) out_L[0] = (float)L;
  if (b >= B) return;
  const unsigned long long* r = red + (size_t)b * 8;
  bool any = (r[5] != 0);
  long long m0 = any ? sunmap(r[0]) : 0;
  long long m1 = any ? sunmap(r[1]) : 0;
  long long m2 = any ? sunmap(r[2]) : 0;
  long long x1 = any ? (sunmap(r[3]) - m1) : 0;
  long long x2 = any ? (sunmap(r[4]) - m2) : 0;
  long long* f = fin + (size_t)b * 8;
  f[0] = m0; f[1] = m1; f[2] = m2; f[3] = x1; f[4] = x2;
}

// ---- kernel 3: build unique 128-bit sort keys (cam | secondary | idx) ------
__global__ void k_keys(const float* __restrict__ lidar_coords,
                       const float* __restrict__ img_coords,
                       const int* __restrict__ lmask,
                       const int* __restrict__ imask,
                       const float* __restrict__ Tinv,
                       const float* __restrict__ Kmat,
                       const long long* __restrict__ fin,
                       ulonglong2* __restrict__ keys,
                       int* __restrict__ padbuf) {
  unsigned t = blockIdx.x * 256 + threadIdx.x;      // [0, B*M)
  if (t >= (unsigned)(B * M)) return;
  unsigned b = t / M, ib = t % M;
  ulonglong2 key;
  if (ib < (unsigned)N) {
    float c[3]; int pad; long long cam;
    project_point((int)b, (int)ib, lidar_coords, img_coords, lmask, imask, Tinv, Kmat,
                  c, &pad, &cam);
    const long long* f = fin + (size_t)b * 8;
    long long q0 = (long long)floorf(c[0] / 16.0f) - f[0];
    long long q1 = (long long)floorf(c[1] / 16.0f) - f[1];
    long long q2 = (long long)floorf(c[2] / 16.0f) - f[2];
    long long s1 = ((q0 & 1) == 1)        ? (f[3] - q1) : q1;
    long long s2 = (((q0 + q1) & 1) == 1) ? (f[4] - q2) : q2;
    long long k64 = q0 * KMULT2 + s1 * KMULT + s2;
    if (pad) k64 = 0x7FFFFFFFFFFFFFFFLL;
    long long sec = (cam == 0) ? k64 : (long long)ib;
    unsigned long long us = (unsigned long long)sec;       // < 2^63
    unsigned long long uc = (unsigned long long)cam;       // small, >= 0
    key.x = (uc << 48) | (us >> 16);
    key.y = (us << 48) | (unsigned long long)ib;           // idx: 15 bits
    padbuf[(size_t)b * N + ib] = pad ? 1 : 0;
  } else {
    key.x = ~0ULL; key.y = ~0ULL;                          // sentinel -> end
  }
  keys[t] = key;
}

// ---- bitonic sort: LDS kernel covering all k = 2..2048 ---------------------
__global__ void k_bitonic_local_full(ulonglong2* __restrict__ keys) {
  __shared__ ulonglong2 sh[2048];
  const unsigned t = threadIdx.x;                 // 1024 threads
  const size_t base = (size_t)blockIdx.x * 2048;
  const unsigned ibBase = (unsigned)(base % M);   // within-batch chunk base
  sh[t]        = keys[base + t];
  sh[t + 1024] = keys[base + t + 1024];
  __syncthreads();
  for (unsigned kk = 2; kk <= 2048; kk <<= 1) {
    for (unsigned j = kk >> 1; j > 0; j >>= 1) {
      unsigned i = ((t & ~(j - 1)) << 1) | (t & (j - 1));
      unsigned l = i | j;
      bool up = (((ibBase + i) & kk) == 0);
      ulonglong2 a = sh[i], bb = sh[l];
      if (key_gt(a, bb) == up) { sh[i] = bb; sh[l] = a; }
      __syncthreads();
    }
  }
  keys[base + t]        = sh[t];
  keys[base + t + 1024] = sh[t + 1024];
}

// ---- bitonic sort: LDS merge kernel (j = 1024..1 of stage kk >= 4096) ------
__global__ void k_bitonic_local_merge(ulonglong2* __restrict__ keys, unsigned kk) {
  __shared__ ulonglong2 sh[2048];
  const unsigned t = threadIdx.x;
  const size_t base = (size_t)blockIdx.x * 2048;
  const unsigned ibBase = (unsigned)(base % M);
  const bool up = ((ibBase & kk) == 0);           // constant per chunk
  sh[t]        = keys[base + t];
  sh[t + 1024] = keys[base + t + 1024];
  __syncthreads();
  for (unsigned j = 1024; j > 0; j >>= 1) {
    unsigned i = ((t & ~(j - 1)) << 1) | (t & (j - 1));
    unsigned l = i | j;
    ulonglong2 a = sh[i], bb = sh[l];
    if (key_gt(a, bb) == up) { sh[i] = bb; sh[l] = a; }
    __syncthreads();
  }
  keys[base + t]        = sh[t];
  keys[base + t + 1024] = sh[t + 1024];
}

// ---- bitonic sort: global pass (stride j >= 2048) --------------------------
__global__ void k_bitonic_global(ulonglong2* __restrict__ keys, unsigned kk, unsigned j) {
  unsigned t = blockIdx.x * 256 + threadIdx.x;    // [0, B*M)
  unsigned b = t / M, ib = t % M;
  unsigned l = ib ^ j;
  if (l > ib) {
    size_t ia = (size_t)b * M + ib, il = (size_t)b * M + l;
    ulonglong2 a = keys[ia], c = keys[il];
    bool up = ((ib & kk) == 0);
    if (key_gt(a, c) == up) { keys[ia] = c; keys[il] = a; }
  }
}

// ---- gather: one wave32 per 1 KB token row; async global<->LDS DMA ---------
__global__ void k_gather(const float* __restrict__ lidar_tokens,
                         const float* __restrict__ img_tokens,
                         const ulonglong2* __restrict__ keys,
                         const int* __restrict__ padbuf,
                         float* __restrict__ out_tokens,
                         float* __restrict__ out_order,
                         float* __restrict__ out_pad) {
  const int wave = threadIdx.x >> 5;
  const int lane = threadIdx.x & 31;
  const unsigned r = blockIdx.x * 8u + (unsigned)wave;     // grid == B*N/8 exactly
  const unsigned b = r / N;
  const unsigned n = r % N;
  ulonglong2 key = keys[(size_t)b * M + n];
  const unsigned src = (unsigned)(key.y & 0xFFFFULL);
  const float* srow = (src < (unsigned)L)
      ? lidar_tokens + ((size_t)b * L + src) * C
      : img_tokens + ((size_t)b * NI + (src - L)) * C;
  float* drow = out_tokens + ((size_t)b * N + n) * C;
#if defined(HAVE_ASYNC_LDS)
  __shared__ __align__(16) float lds[8 * C];               // 1 KB per wave
  float* lrow = &lds[wave * C];
  {
    gv4p g0 = (gv4p)(void*)(srow + lane * 4);
    gv4p g1 = (gv4p)(void*)(srow + 128 + lane * 4);
    lv4p l0 = (lv4p)(void*)(lrow + lane * 4);
    lv4p l1 = (lv4p)(void*)(lrow + 128 + lane * 4);
    __builtin_amdgcn_global_load_async_to_lds_b128(g0, l0, 0, 0);
    __builtin_amdgcn_global_load_async_to_lds_b128(g1, l1, 0, 0);
    wait_asynccnt0();                                      // LDS writes visible
    __builtin_amdgcn_global_store_async_from_lds_b128((gv4p)(void*)(drow + lane * 4), l0, 0, 0);
    __builtin_amdgcn_global_store_async_from_lds_b128((gv4p)(void*)(drow + 128 + lane * 4), l1, 0, 0);
    wait_asynccnt0();                                      // LDS reusable
  }
#else
  const float4* s4 = (const float4*)srow;
  float4* d4 = (float4*)drow;
  d4[lane]      = s4[lane];
  d4[lane + 32] = s4[lane + 32];
#endif
  if (lane == 0) {
    out_order[(size_t)b * N + n] = (float)src;
    out_pad[(size_t)b * N + n]   = (float)(padbuf[(size_t)b * N + src] & 1);
  }
}

// ---------------------------------------------------------------------------
extern "C" void kernel_launch(void* const* d_in, const int* in_sizes, int n_in,
                              void* d_out, int out_size, void* d_ws, size_t ws_size,
                              hipStream_t stream) {
  (void)in_sizes; (void)n_in; (void)out_size; (void)ws_size;
  const float* lidar_tokens = (const float*)d_in[0];
  const float* lidar_coords = (const float*)d_in[1];
  const float* img_tokens   = (const float*)d_in[2];
  const float* img_coords   = (const float*)d_in[3];
  const float* Kmat         = (const float*)d_in[4];
  const float* T_c2w        = (const float*)d_in[5];
  const int*   lmask        = (const int*)d_in[6];
  const int*   imask        = (const int*)d_in[7];

  char* ws = (char*)d_ws;
  float*              Tinv   = (float*)(ws + 0);                       // 512 B
  unsigned long long* red    = (unsigned long long*)(ws + 512);        // 512 B
  long long*          fin    = (long long*)(ws + 1024);                // 512 B
  int*                padbuf = (int*)(ws + 1536);                      // B*N*4
  ulonglong2*         keys   = (ulonglong2*)(ws + 1536 + (size_t)B * N * 4); // B*M*16

  float* out_tokens = (float*)d_out;
  float* out_order  = out_tokens + (size_t)B * N * C;
  float* out_pad    = out_order + (size_t)B * N;
  float* out_L      = out_pad + (size_t)B * N;

  k_init<<<1, 64, 0, stream>>>(T_c2w, Tinv, red);
  k_reduce<<<(B * N + 255) / 256, 256, 0, stream>>>(lidar_coords, img_coords, lmask,
                                                    imask, Tinv, Kmat, red);
  k_finalize<<<1, 64, 0, stream>>>(red, fin, out_L);
  k_keys<<<(B * M + 255) / 256, 256, 0, stream>>>(lidar_coords, img_coords, lmask,
                                                  imask, Tinv, Kmat, fin, keys, padbuf);
  k_bitonic_local_full<<<(B * M) / 2048, 1024, 0, stream>>>(keys);
  for (unsigned kk = 4096; kk <= 32768u; kk <<= 1) {
    for (unsigned j = kk >> 1; j >= 2048; j >>= 1)
      k_bitonic_global<<<(B * M) / 256, 256, 0, stream>>>(keys, kk, j);
    k_bitonic_local_merge<<<(B * M) / 2048, 1024, 0, stream>>>(keys, kk);
  }
  k_gather<<<(B * N) / 8, 256, 0, stream>>>(lidar_tokens, img_tokens, keys, padbuf,
                                            out_tokens, out_order, out_pad);
}